// SpecialSpmmFinal_e2t_37641093382713
// MI455X (gfx1250) — compile-verified
//
#include <hip/hip_runtime.h>
#include <cstdint>

// segment_sum(edge_w[E,64], edge[0], N1) -> out[N1,64]
// Memory-bound scatter-add. HBM floor: 256MB NT stream read + ~25.6MB output
// writeback ~= 12us @ 23.3 TB/s. 64M f32 atomics resolve in L2 (output fits
// in 192MB L2; input stream is marked non-temporal so it can't evict it).
// Pipeline: double-buffered async global->LDS staging (ASYNCcnt), then
// LDS -> native global_atomic_add_f32 scatter.

#define D_FEAT        64
#define THREADS       256
#define STAGE_EDGES   128                                  // edges per stage
#define STAGES_PER_BLOCK 8
#define EDGES_PER_BLOCK (STAGE_EDGES * STAGES_PER_BLOCK)   // 1024
#define CHUNKS_PER_STAGE (STAGE_EDGES * (D_FEAT / 4))      // 2048 float4 chunks
#define CHUNKS_PER_THREAD (CHUNKS_PER_STAGE / THREADS)     // 8 async loads/thread/stage

__global__ void zero_out_kernel(float4* __restrict__ out4, int n4) {
  int i = blockIdx.x * blockDim.x + threadIdx.x;
  int stride = gridDim.x * blockDim.x;
  float4 z; z.x = 0.f; z.y = 0.f; z.z = 0.f; z.w = 0.f;
  for (; i < n4; i += stride) out4[i] = z;
}

__global__ void __launch_bounds__(THREADS)
scatter_add_kernel(const int* __restrict__ row,
                   const float* __restrict__ w,
                   float* __restrict__ out,
                   int E) {
  // 2 x 32KB double buffer (64KB total; WGP has 320KB LDS)
  __shared__ float buf[2][STAGE_EDGES * D_FEAT];

  const int t = threadIdx.x;
  const long long blockBase = (long long)blockIdx.x * EDGES_PER_BLOCK;
  const unsigned long long wbase = (unsigned long long)(uintptr_t)w;   // SGPR base
  const unsigned long long obase = (unsigned long long)(uintptr_t)out; // SGPR base
  const unsigned wlimit = (unsigned)E * (D_FEAT * 4u) - 16u;  // last valid 16B chunk

  // Issue one stage of async global->LDS B128 loads (GVS form: SGPR base +
  // 32-bit lane offset; input is <256MB so offsets fit u32). Within a stage
  // the stream is contiguous: chunk c sits at stage_base + c*16 in memory
  // AND at lbase + c*16 in LDS. Every thread issues exactly CHUNKS_PER_THREAD
  // instructions (offsets clamped in the tail) so per-wave ASYNCcnt counts
  // are uniform and the s_wait_asynccnt immediates are exact.
  auto issue_stage = [&](int stage, int bufIdx) {
    const unsigned stageByteBase =
        (unsigned)(blockBase + (long long)stage * STAGE_EDGES) * (D_FEAT * 4u);
    // low 32 bits of a generic shared pointer == LDS byte offset
    const unsigned lbase =
        (unsigned)(unsigned long long)(uintptr_t)&buf[bufIdx][0];
#pragma unroll
    for (int k = 0; k < CHUNKS_PER_THREAD; ++k) {
      const int c = t + k * THREADS;          // chunk id in [0, 2048)
      unsigned goff = stageByteBase + (unsigned)c * 16u;
      if (goff > wlimit) goff = wlimit;       // clamp: keep addr valid in tail
      const unsigned la = lbase + (unsigned)c * 16u;
      // non-temporal: 256MB single-use stream must not evict the L2-resident
      // output lines that the atomics below RMW against
      asm volatile("global_load_async_to_lds_b128 %0, %1, %2 th:TH_LOAD_NT"
                   :: "v"(la), "v"(goff), "s"(wbase)
                   : "memory");
    }
  };

  issue_stage(0, 0);

  for (int s = 0; s < STAGES_PER_BLOCK; ++s) {
    if (s + 1 < STAGES_PER_BLOCK) {
      issue_stage(s + 1, (s + 1) & 1);
      // async loads complete in order: <=8 outstanding => stage s fully in LDS
      asm volatile("s_wait_asynccnt 8" ::: "memory");
    } else {
      asm volatile("s_wait_asynccnt 0" ::: "memory");
    }
    __syncthreads();  // all 8 waves' stage-s data visible in LDS

    const long long e0 = blockBase + (long long)s * STAGE_EDGES;
    const float* lb = &buf[s & 1][0];
#pragma unroll
    for (int k = 0; k < CHUNKS_PER_THREAD; ++k) {
      const int c = t + k * THREADS;
      const long long e = e0 + (c >> 4);      // 16 float4-chunks per edge
      if (e < (long long)E) {
        const int r = row[e];
        const float4 v = *reinterpret_cast<const float4*>(lb + (size_t)c * 4);
        // output byte offset: r*256 + (c&15)*16  (max 25.6MB, fits u32)
        const unsigned ooff = ((unsigned)r << 8) + ((unsigned)(c & 15) << 4);
        // native no-return f32 atomic adds, executed by the L2 atomic units
        asm volatile("global_atomic_add_f32 %0, %1, %2"
                     :: "v"(ooff), "v"(v.x), "s"(obase) : "memory");
        asm volatile("global_atomic_add_f32 %0, %1, %2 offset:4"
                     :: "v"(ooff), "v"(v.y), "s"(obase) : "memory");
        asm volatile("global_atomic_add_f32 %0, %1, %2 offset:8"
                     :: "v"(ooff), "v"(v.z), "s"(obase) : "memory");
        asm volatile("global_atomic_add_f32 %0, %1, %2 offset:12"
                     :: "v"(ooff), "v"(v.w), "s"(obase) : "memory");
      }
    }
    __syncthreads();  // buffer (s&1) free for reuse by stage s+2
  }
}

extern "C" void kernel_launch(void* const* d_in, const int* in_sizes, int n_in,
                              void* d_out, int out_size, void* d_ws, size_t ws_size,
                              hipStream_t stream) {
  const int* edge = (const int*)d_in[0];       // [2, E] int32; edge[0..E) = rows
  const float* edge_w = (const float*)d_in[1]; // [E, 64] f32
  const int E = in_sizes[1] / D_FEAT;          // 1,000,000
  const int* row = edge;                       // first E entries

  float* out = (float*)d_out;                  // [N1, 64] f32
  const int n4 = out_size / 4;

  // 1) zero the accumulator (harness poisons d_out); regular-temporal stores
  //    also pull the output lines into L2 dirty, priming them for the atomics
  int zblocks = (n4 + THREADS - 1) / THREADS;
  if (zblocks > 4096) zblocks = 4096;
  zero_out_kernel<<<zblocks, THREADS, 0, stream>>>((float4*)out, n4);

  // 2) async-staged scatter-add
  const int blocks = (E + EDGES_PER_BLOCK - 1) / EDGES_PER_BLOCK;  // 977
  scatter_add_kernel<<<blocks, THREADS, 0, stream>>>(row, edge_w, out, E);
}